// PointNet2SSGSemanticSegmentation_18837726561046
// MI455X (gfx1250) — compile-verified
//
#include <hip/hip_runtime.h>
#include <hip/hip_bf16.h>
#include <math.h>

// ---------------------------------------------------------------------------
// PointNet++ SSG semantic segmentation forward pass for gfx1250 (MI455X).
// wave32 everywhere; MLP GEMMs use V_WMMA_F32_16X16X4_F32 with a branch-free
// inner loop (padded operands, 32x32 macro-tile per wave, 4 WMMAs / 4 b64
// loads per K-step).
// ---------------------------------------------------------------------------

typedef float v2f __attribute__((ext_vector_type(2)));
typedef float v8f __attribute__((ext_vector_type(8)));

static constexpr int BATCH = 8;
static constexpr int NPTS  = 4096;
static constexpr int THREADS = 256;

// -------------------------------- kernels ----------------------------------

__global__ void k_zero(float* __restrict__ p, int n) {
  int t = blockIdx.x * blockDim.x + threadIdx.x;
  if (t < n) p[t] = 0.0f;
}

// Zero-pad weights: W[C x K] -> Wp[Cpad x Kp] (rows >= C and cols >= K are 0).
__global__ void k_pad_w(const float* __restrict__ W, int C, int K,
                        float* __restrict__ Wp, int Kp, int total) {
  int t = blockIdx.x * blockDim.x + threadIdx.x;
  if (t >= total) return;
  int r = t / Kp, k = t % Kp;
  Wp[t] = (r < C && k < K) ? W[(size_t)r * K + k] : 0.0f;
}

// (B,9,N) -> xyz (B,N,3), pts (B,N,6)
__global__ void k_split_transpose(const float* __restrict__ in,
                                  float* __restrict__ xyz,
                                  float* __restrict__ pts, int N, int total) {
  int t = blockIdx.x * blockDim.x + threadIdx.x;
  if (t >= total) return;
  int b = t / N, n = t % N;
  const float* src = in + (size_t)b * 9 * N + n;
  float* x = xyz + (size_t)t * 3;
  float* p = pts + (size_t)t * 6;
  x[0] = src[0]; x[1] = src[(size_t)N]; x[2] = src[(size_t)2 * N];
#pragma unroll
  for (int c = 0; c < 6; c++) p[c] = src[(size_t)(3 + c) * N];
}

// Farthest point sampling: one block per batch, serial npoint loop,
// parallel over N with LDS argmax reduction (ties -> lower index, like argmax).
__global__ void k_fps(const float* __restrict__ xyz, int N, int npoint,
                      int* __restrict__ fidx, float* __restrict__ distws) {
  int b = blockIdx.x;
  xyz    += (size_t)b * N * 3;
  fidx   += (size_t)b * npoint;
  float* dist = distws + (size_t)b * N;
  int tid = threadIdx.x;
  for (int i = tid; i < N; i += THREADS) dist[i] = 1e10f;
  __shared__ float sval[THREADS];
  __shared__ int   sidx[THREADS];
  __syncthreads();
  int far = 0;
  for (int s = 0; s < npoint; s++) {
    if (tid == 0) fidx[s] = far;
    float fx = xyz[far * 3 + 0], fy = xyz[far * 3 + 1], fz = xyz[far * 3 + 2];
    float best = -1.0f; int bi = 0;
    for (int i = tid; i < N; i += THREADS) {
      float dx = xyz[i * 3 + 0] - fx;
      float dy = xyz[i * 3 + 1] - fy;
      float dz = xyz[i * 3 + 2] - fz;
      float d  = dx * dx + dy * dy + dz * dz;
      float dd = fminf(dist[i], d);
      dist[i]  = dd;
      if (dd > best) { best = dd; bi = i; }
    }
    sval[tid] = best; sidx[tid] = bi;
    __syncthreads();
    for (int off = THREADS / 2; off > 0; off >>= 1) {
      if (tid < off) {
        float ov = sval[tid + off]; int oi = sidx[tid + off];
        if (ov > sval[tid] || (ov == sval[tid] && oi < sidx[tid])) {
          sval[tid] = ov; sidx[tid] = oi;
        }
      }
      __syncthreads();
    }
    far = sidx[0];
    __syncthreads();
  }
}

// dst[b,s,c] = src[b, idx[b,s], c]
__global__ void k_gather_rows(const float* __restrict__ src,
                              const int* __restrict__ idx,
                              int Nout, int Nin, int C,
                              float* __restrict__ dst, int total) {
  int t = blockIdx.x * blockDim.x + threadIdx.x;
  if (t >= total) return;
  int b = t / (Nout * C), r = t % (Nout * C);
  int s = r / C, c = r % C;
  dst[t] = src[((size_t)b * Nin + idx[b * Nout + s]) * C + c];
}

// Ball query + grouping: one wave32 per center. Selects the first `nsample`
// in-radius points in index order via ballot compaction; pads with the first
// hit (matches sort-based reference). Writes rows [dx,dy,dz, pts..., 0-pad]
// of stride Ks (the padded GEMM K).
__global__ void k_group(const float* __restrict__ xyz, const float* __restrict__ pts,
                        const float* __restrict__ new_xyz, int N, int npoint,
                        int nsample, int Cpts, int Ks, float r2,
                        float* __restrict__ out) {
  int center = blockIdx.x;               // b*npoint + s
  int b    = center / npoint;
  int lane = threadIdx.x;                // 0..31 (wave32 block)
  const float* nc = new_xyz + (size_t)center * 3;
  float cx = nc[0], cy = nc[1], cz = nc[2];
  const float* X = xyz + (size_t)b * N * 3;
  const float* P = pts + (size_t)b * N * Cpts;
  float* orow = out + (size_t)center * nsample * Ks;
  int count = 0, firstIdx = -1;
  for (int base = 0; base < N && count < nsample; base += 32) {
    int i = base + lane;
    float dx = 0.f, dy = 0.f, dz = 0.f;
    bool ok = false;
    if (i < N) {
      dx = X[i * 3 + 0] - cx; dy = X[i * 3 + 1] - cy; dz = X[i * 3 + 2] - cz;
      ok = (dx * dx + dy * dy + dz * dz) <= r2;
    }
    unsigned mask = (unsigned)__ballot(ok);     // wave32: low 32 bits
    if (ok) {
      int slot = count + __popc(mask & ((1u << lane) - 1u));
      if (slot < nsample) {
        float* row = orow + (size_t)slot * Ks;
        row[0] = dx; row[1] = dy; row[2] = dz;
        for (int c = 0; c < Cpts; c++) row[3 + c] = P[(size_t)i * Cpts + c];
        for (int c = 3 + Cpts; c < Ks; c++) row[c] = 0.0f;
      }
    }
    if (firstIdx < 0 && mask) firstIdx = base + __ffs(mask) - 1;
    count += __popc(mask);
  }
  if (count > nsample) count = nsample;
  if (count < nsample) {                        // pad with first in-radius point
    int fi = (firstIdx >= 0) ? firstIdx : 0;
    float dx = X[fi * 3 + 0] - cx, dy = X[fi * 3 + 1] - cy, dz = X[fi * 3 + 2] - cz;
    for (int slot = count + lane; slot < nsample; slot += 32) {
      float* row = orow + (size_t)slot * Ks;
      row[0] = dx; row[1] = dy; row[2] = dz;
      for (int c = 0; c < Cpts; c++) row[3 + c] = P[(size_t)fi * Cpts + c];
      for (int c = 3 + Cpts; c < Ks; c++) row[c] = 0.0f;
    }
  }
}

// GEMM Y[M,C] = X[M,Ks] * Wp[Cpad,Ks]^T via V_WMMA_F32_16X16X4_F32.
// Requirements (guaranteed by construction): M % 32 == 0, Ks % 4 == 0,
// Cpad % 32 == 0, X rows have stride Ks with zeros in pad columns, Wp is
// zero-padded.  One wave owns a 32x32 output macro-tile: 4 accumulators,
// A/B fragments each reused twice -> branch-free inner loop.
// fp32 fragment layout (ISA 7.12.2): lanes 0-15 hold K=k0,k0+1; lanes 16-31
// hold K=k0+2,k0+3.  D: acc[v] = (m = half*8+v, n = lane&15).
// Also accumulates per-channel sum/sumsq (whole-tensor BN) when doStats != 0.
__global__ void k_gemm(const float* __restrict__ X, const float* __restrict__ Wp,
                       float* __restrict__ Y, int M, int Ks, int C, int Cpad,
                       float* __restrict__ sums, float* __restrict__ sumsqs,
                       int doStats) {
  int wave = threadIdx.x >> 5;
  int lane = threadIdx.x & 31;
  int nCt = Cpad >> 5;
  int nMt = M >> 5;
  int tile = blockIdx.x * 8 + wave;          // 256 threads = 8 waves/block
  if (tile >= nCt * nMt) return;             // wave-uniform exit (EXEC stays all-1)
  int tm = tile / nCt, tn = tile % nCt;
  int m0 = tm << 5, n0 = tn << 5;
  int half = lane >> 4;
  int l    = lane & 15;
  const float* ap0 = X  + (size_t)(m0 + l) * Ks + half * 2;
  const float* ap1 = ap0 + (size_t)16 * Ks;
  const float* bp0 = Wp + (size_t)(n0 + l) * Ks + half * 2;
  const float* bp1 = bp0 + (size_t)16 * Ks;
  v8f acc00 = {}, acc01 = {}, acc10 = {}, acc11 = {};
#pragma unroll 2
  for (int k0 = 0; k0 < Ks; k0 += 4) {
    v2f a0 = *(const v2f*)(ap0 + k0);
    v2f a1 = *(const v2f*)(ap1 + k0);
    v2f b0 = *(const v2f*)(bp0 + k0);
    v2f b1 = *(const v2f*)(bp1 + k0);
    acc00 = __builtin_amdgcn_wmma_f32_16x16x4_f32(false, a0, false, b0, (short)0, acc00, false, false);
    acc01 = __builtin_amdgcn_wmma_f32_16x16x4_f32(false, a0, false, b1, (short)0, acc01, false, false);
    acc10 = __builtin_amdgcn_wmma_f32_16x16x4_f32(false, a1, false, b0, (short)0, acc10, false, false);
    acc11 = __builtin_amdgcn_wmma_f32_16x16x4_f32(false, a1, false, b1, (short)0, acc11, false, false);
  }
  int na = n0 + l, nb = na + 16;
  float s_a = 0.f, s2_a = 0.f, s_b = 0.f, s2_b = 0.f;
#pragma unroll
  for (int v = 0; v < 8; v++) {
    int mA = m0 + half * 8 + v;
    int mB = mA + 16;
    if (na < C) {
      float x0 = acc00[v]; Y[(size_t)mA * C + na] = x0; s_a += x0; s2_a += x0 * x0;
      float x1 = acc10[v]; Y[(size_t)mB * C + na] = x1; s_a += x1; s2_a += x1 * x1;
    }
    if (nb < C) {
      float x0 = acc01[v]; Y[(size_t)mA * C + nb] = x0; s_b += x0; s2_b += x0 * x0;
      float x1 = acc11[v]; Y[(size_t)mB * C + nb] = x1; s_b += x1; s2_b += x1 * x1;
    }
  }
  if (doStats) {
    if (na < C) { atomicAdd(&sums[na], s_a); atomicAdd(&sumsqs[na], s2_a); }
    if (nb < C) { atomicAdd(&sums[nb], s_b); atomicAdd(&sumsqs[nb], s2_b); }
  }
}

// y = relu((y-mean)*rsqrt(var+1e-5)*g + b), stats over all M rows per channel.
__global__ void k_bn_relu(float* __restrict__ Y, int M, int C,
                          const float* __restrict__ sums,
                          const float* __restrict__ sumsqs,
                          const float* __restrict__ g,
                          const float* __restrict__ bta, int total) {
  int t = blockIdx.x * blockDim.x + threadIdx.x;
  if (t >= total) return;
  int n = t % C;
  float invM = 1.0f / (float)M;
  float mean = sums[n] * invM;
  float var  = sumsqs[n] * invM - mean * mean;
  float x = (Y[t] - mean) * rsqrtf(var + 1e-5f) * g[n] + bta[n];
  Y[t] = fmaxf(x, 0.0f);
}

// (rows, ns, C) -> max over ns -> (rows, C)
__global__ void k_maxpool(const float* __restrict__ X, int rows, int ns, int C,
                          float* __restrict__ out, int total) {
  int t = blockIdx.x * blockDim.x + threadIdx.x;
  if (t >= total) return;
  int r = t / C, c = t % C;
  const float* p = X + ((size_t)r * ns) * C + c;
  float m = p[0];
  for (int j = 1; j < ns; j++) m = fmaxf(m, p[(size_t)j * C]);
  out[t] = m;
}

// 3-NN inverse-distance interpolation + concat: row = [p1 (C1) | interp (C2) | 0-pad],
// row stride Ks (padded GEMM K).
__global__ void k_fp_knn_concat(const float* __restrict__ xyz1,
                                const float* __restrict__ xyz2,
                                const float* __restrict__ p1,
                                const float* __restrict__ p2,
                                int N1, int N2, int C1, int C2, int Ks,
                                float* __restrict__ Xout, int total) {
  int t = blockIdx.x * blockDim.x + threadIdx.x;
  if (t >= total) return;                       // total = B*N1
  int b = t / N1;
  const float* q = xyz1 + (size_t)t * 3;
  float qx = q[0], qy = q[1], qz = q[2];
  const float* ref = xyz2 + (size_t)b * N2 * 3;
  float d0 = 3.4e38f, d1 = 3.4e38f, d2 = 3.4e38f;
  int   i0 = 0, i1 = 0, i2 = 0;
  for (int j = 0; j < N2; j++) {
    float dx = ref[j * 3 + 0] - qx;
    float dy = ref[j * 3 + 1] - qy;
    float dz = ref[j * 3 + 2] - qz;
    float d  = dx * dx + dy * dy + dz * dz;
    if (d < d0)      { d2 = d1; i2 = i1; d1 = d0; i1 = i0; d0 = d; i0 = j; }
    else if (d < d1) { d2 = d1; i2 = i1; d1 = d;  i1 = j; }
    else if (d < d2) { d2 = d;  i2 = j; }
  }
  float w0 = 1.f / (d0 + 1e-8f), w1 = 1.f / (d1 + 1e-8f), w2 = 1.f / (d2 + 1e-8f);
  float wsum = w0 + w1 + w2;
  w0 /= wsum; w1 /= wsum; w2 /= wsum;
  float* row = Xout + (size_t)t * Ks;
  const float* pr = p1 + (size_t)t * C1;
  for (int c = 0; c < C1; c++) row[c] = pr[c];
  const float* a0 = p2 + ((size_t)b * N2 + i0) * C2;
  const float* a1 = p2 + ((size_t)b * N2 + i1) * C2;
  const float* a2 = p2 + ((size_t)b * N2 + i2) * C2;
  for (int c = 0; c < C2; c++)
    row[C1 + c] = w0 * a0[c] + w1 * a1[c] + w2 * a2[c];
  for (int c = C1 + C2; c < Ks; c++) row[c] = 0.0f;
}

// out[b,c,n] = Y[(b*N+n), c] + bias[c]   (head: no BN/ReLU)
__global__ void k_bias_out(const float* __restrict__ Y, const float* __restrict__ bias,
                           int N, int C, float* __restrict__ out, int total) {
  int t = blockIdx.x * blockDim.x + threadIdx.x;
  if (t >= total) return;
  int b = t / (C * N), r = t % (C * N);
  int c = r / N, n = r % N;
  out[t] = Y[((size_t)b * N + n) * C + c] + bias[c];
}

// ------------------------------ host helpers -------------------------------

struct Layer {
  const float* Wp;  // padded weights [Cpad x Kp]
  const float* g;
  const float* b;
  int C;     // true output channels
  int Cpad;  // C rounded up to 32
  int Kp;    // input K rounded up to 4 (== producer row stride)
};

static inline int cdiv(int a, int b) { return (a + b - 1) / b; }

static void launch_layer(const float* X, float* Y, int M, const Layer& L,
                         float* stats, hipStream_t st) {
  k_zero<<<cdiv(1024, THREADS), THREADS, 0, st>>>(stats, 1024);
  int tiles = (M >> 5) * (L.Cpad >> 5);
  k_gemm<<<cdiv(tiles, 8), THREADS, 0, st>>>(X, L.Wp, Y, M, L.Kp, L.C, L.Cpad,
                                             stats, stats + 512, 1);
  k_bn_relu<<<cdiv(M * L.C, THREADS), THREADS, 0, st>>>(Y, M, L.C, stats,
                                                        stats + 512, L.g, L.b,
                                                        M * L.C);
}

// Ping-pongs between bufA/bufB; last layer goes to outBuf if non-null.
static const float* run_mlp(const float* X0, float* bufA, float* bufB,
                            float* outBuf, int M, const Layer* Ls, int nl,
                            float* stats, hipStream_t st) {
  const float* cur = X0;
  for (int i = 0; i < nl; i++) {
    float* dst = (i == nl - 1 && outBuf) ? outBuf
                                         : ((cur == bufA) ? bufB : bufA);
    launch_layer(cur, dst, M, Ls[i], stats, st);
    cur = dst;
  }
  return cur;
}

static void sa_stage(const float* xyz_in, const float* pts_in, int Nin,
                     int npoint, float radius, int nsample, int Cpts,
                     const Layer* Ls, int nl, float* xyz_out, float* feat_out,
                     int* fidx, float* distws, float* bufA, float* bufB,
                     float* stats, hipStream_t st) {
  k_fps<<<BATCH, THREADS, 0, st>>>(xyz_in, Nin, npoint, fidx, distws);
  int tg = BATCH * npoint * 3;
  k_gather_rows<<<cdiv(tg, THREADS), THREADS, 0, st>>>(xyz_in, fidx, npoint,
                                                       Nin, 3, xyz_out, tg);
  k_group<<<BATCH * npoint, 32, 0, st>>>(xyz_in, pts_in, xyz_out, Nin, npoint,
                                         nsample, Cpts, Ls[0].Kp,
                                         radius * radius, bufA);
  int M = BATCH * npoint * nsample;
  const float* fin = run_mlp(bufA, bufA, bufB, nullptr, M, Ls, nl, stats, st);
  int Cl = Ls[nl - 1].C;
  int tm = BATCH * npoint * Cl;
  k_maxpool<<<cdiv(tm, THREADS), THREADS, 0, st>>>(fin, BATCH * npoint,
                                                   nsample, Cl, feat_out, tm);
}

static void fp_stage(const float* xyz1, int N1, const float* xyz2, int N2,
                     const float* p1, int C1, const float* p2, int C2,
                     const Layer* Ls, int nl, float* outBuf, float* bufA,
                     float* bufB, float* stats, hipStream_t st) {
  int tk = BATCH * N1;
  k_fp_knn_concat<<<cdiv(tk, 128), 128, 0, st>>>(xyz1, xyz2, p1, p2, N1, N2,
                                                 C1, C2, Ls[0].Kp, bufA, tk);
  run_mlp(bufA, bufA, bufB, outBuf, BATCH * N1, Ls, nl, stats, st);
}

// ------------------------------- entry point -------------------------------

extern "C" void kernel_launch(void* const* d_in, const int* in_sizes, int n_in,
                              void* d_out, int out_size, void* d_ws,
                              size_t ws_size, hipStream_t stream) {
  (void)in_sizes; (void)n_in; (void)out_size; (void)ws_size;
  auto P = [&](int i) { return (const float*)d_in[i]; };

  // Workspace carve-up (floats, 64-elt aligned).
  float* w = (float*)d_ws;
  size_t off = 0;
  auto alloc = [&](size_t n) {
    float* p = w + off;
    off += (n + 63) & ~(size_t)63;
    return p;
  };

  // Params in setup_inputs() insertion order:
  // inputs, sa1..sa4 (3x W,g,b each), fp4..fp1, out (W,g,b), head (W,b).
  // Each layer's weights are copied to a zero-padded [Cpad32 x Kp4] buffer so
  // the WMMA GEMM inner loop is branch-free.
  auto mkLayer = [&](int wi, int C, int K) {
    int Cp = (C + 31) & ~31;
    int Kp = (K + 3) & ~3;
    float* Wp = alloc((size_t)Cp * Kp);
    int tot = Cp * Kp;
    k_pad_w<<<cdiv(tot, THREADS), THREADS, 0, stream>>>(P(wi), C, K, Wp, Kp, tot);
    Layer L = {Wp, P(wi + 1), P(wi + 2), C, Cp, Kp};
    return L;
  };

  Layer sa1L[3] = {mkLayer(1, 32, 9),    mkLayer(4, 32, 32),   mkLayer(7, 64, 32)};
  Layer sa2L[3] = {mkLayer(10, 64, 67),  mkLayer(13, 64, 64),  mkLayer(16, 128, 64)};
  Layer sa3L[3] = {mkLayer(19, 128, 131),mkLayer(22, 128, 128),mkLayer(25, 256, 128)};
  Layer sa4L[3] = {mkLayer(28, 256, 259),mkLayer(31, 256, 256),mkLayer(34, 512, 256)};
  Layer fp4L[2] = {mkLayer(37, 256, 768),mkLayer(40, 256, 256)};
  Layer fp3L[2] = {mkLayer(43, 256, 384),mkLayer(46, 256, 256)};
  Layer fp2L[2] = {mkLayer(49, 256, 320),mkLayer(52, 128, 256)};
  Layer fp1L[3] = {mkLayer(55, 128, 134),mkLayer(58, 128, 128),mkLayer(61, 128, 128)};
  Layer outL[1] = {mkLayer(64, 128, 128)};
  // head: linear only (no BN), C=13 -> Cpad=32
  int headCp = 32, headKp = 128;
  float* headWp = alloc((size_t)headCp * headKp);
  {
    int tot = headCp * headKp;
    k_pad_w<<<cdiv(tot, THREADS), THREADS, 0, stream>>>(P(67), 13, 128, headWp,
                                                        headKp, tot);
  }
  const float* headB = P(68);

  float* xyz0   = alloc((size_t)BATCH * NPTS * 3);
  float* pts0   = alloc((size_t)BATCH * NPTS * 6);
  float* distws = alloc((size_t)BATCH * NPTS);
  int*   fidx   = (int*)alloc((size_t)BATCH * 1024);
  float* l1x = alloc((size_t)BATCH * 1024 * 3);
  float* l1p = alloc((size_t)BATCH * 1024 * 64);
  float* l2x = alloc((size_t)BATCH * 256 * 3);
  float* l2p = alloc((size_t)BATCH * 256 * 128);
  float* l3x = alloc((size_t)BATCH * 64 * 3);
  float* l3p = alloc((size_t)BATCH * 64 * 256);
  float* l4x = alloc((size_t)BATCH * 16 * 3);
  float* l4p = alloc((size_t)BATCH * 16 * 512);
  float* fp4o = alloc((size_t)BATCH * 64 * 256);
  float* fp3o = alloc((size_t)BATCH * 256 * 256);
  float* fp2o = alloc((size_t)BATCH * 1024 * 128);
  float* fp1o = alloc((size_t)BATCH * NPTS * 128);
  float* outo = alloc((size_t)BATCH * NPTS * 128);
  float* headY = alloc((size_t)BATCH * NPTS * 13 + 64);
  float* stats = alloc(1024);                       // [sum(512) | sumsq(512)]
  size_t bufElems = (size_t)BATCH * 1024 * 32 * 64; // SA1 grouped tensor, max C
  float* bufA = alloc(bufElems);
  float* bufB = alloc(bufElems);

  hipStream_t st = stream;
  int tsp = BATCH * NPTS;
  k_split_transpose<<<cdiv(tsp, THREADS), THREADS, 0, st>>>((const float*)d_in[0],
                                                            xyz0, pts0, NPTS, tsp);

  // Set abstraction pyramid
  sa_stage(xyz0, pts0, NPTS, 1024, 0.1f, 32, 6,   sa1L, 3, l1x, l1p, fidx, distws, bufA, bufB, stats, st);
  sa_stage(l1x,  l1p,  1024, 256,  0.2f, 32, 64,  sa2L, 3, l2x, l2p, fidx, distws, bufA, bufB, stats, st);
  sa_stage(l2x,  l2p,  256,  64,   0.4f, 32, 128, sa3L, 3, l3x, l3p, fidx, distws, bufA, bufB, stats, st);
  sa_stage(l3x,  l3p,  64,   16,   0.8f, 32, 256, sa4L, 3, l4x, l4p, fidx, distws, bufA, bufB, stats, st);

  // Feature propagation (3-NN interpolation + MLPs)
  fp_stage(l3x, 64,   l4x, 16,   l3p, 256, l4p,  512, fp4L, 2, fp4o, bufA, bufB, stats, st);
  fp_stage(l2x, 256,  l3x, 64,   l2p, 128, fp4o, 256, fp3L, 2, fp3o, bufA, bufB, stats, st);
  fp_stage(l1x, 1024, l2x, 256,  l1p, 64,  fp3o, 256, fp2L, 2, fp2o, bufA, bufB, stats, st);
  fp_stage(xyz0, NPTS, l1x, 1024, pts0, 6, fp2o, 128, fp1L, 3, fp1o, bufA, bufB, stats, st);

  // out MLP (BN+ReLU) then head (linear + bias, transposed to (B,13,N))
  run_mlp(fp1o, bufA, bufB, outo, BATCH * NPTS, outL, 1, stats, st);
  int Mh = BATCH * NPTS;
  int tilesH = (Mh >> 5) * (headCp >> 5);
  k_gemm<<<cdiv(tilesH, 8), THREADS, 0, st>>>(outo, headWp, headY, Mh, headKp,
                                              13, headCp, nullptr, nullptr, 0);
  int to = BATCH * 13 * NPTS;
  k_bias_out<<<cdiv(to, THREADS), THREADS, 0, st>>>(headY, headB, NPTS, 13,
                                                    (float*)d_out, to);
}